// RTMCCLoss_16552803959404
// MI455X (gfx1250) — compile-verified
//
#include <hip/hip_runtime.h>
#include <math.h>

namespace {
constexpr int   kN            = 256;
constexpr int   kK            = 133;
constexpr int   kW            = 512;                 // = H = round(256 * 2.0)
constexpr int   kRowLen       = 512;                 // elements per softmax row
constexpr int   kRows         = kN * kK * 2;         // 68096
constexpr int   kWavesPerBlk  = 8;                   // 256 threads, wave32
constexpr int   kBlocks       = kRows / kWavesPerBlk;// 8512 (exact)
constexpr float kSigma        = 5.66f;
constexpr float kBeta         = 10.0f;               // BETA == LABEL_BETA
}

// One wave32 per 512-element row. Row is DMA'd global->LDS via the CDNA5
// async path (ASYNCcnt), then pulled into 16 registers/lane for the fused
// log-softmax / KL evaluation. Cross-lane reductions are 5 xor-shuffles.
__global__ __launch_bounds__(256) void rtmcc_row_kernel(
    const float* __restrict__ pred,
    const float* __restrict__ kp,
    float* __restrict__ part)
{
  __shared__ __align__(16) float buf[kWavesPerBlk][kRowLen];  // 16 KB
  __shared__ float acc[kWavesPerBlk];

  const int wv   = threadIdx.x >> 5;
  const int lane = threadIdx.x & 31;
  const int row  = blockIdx.x * kWavesPerBlk + wv;

  float rowres = 0.0f;
  if (row < kRows) {
    const int nk   = row >> 1;     // which (n,k) keypoint
    const int half = row & 1;      // 0 = x-row, 1 = y-row

    const float kx  = kp[nk * 3 + 0];
    const float ky  = kp[nk * 3 + 1];
    const float vis = kp[nk * 3 + 2];
    const float mux = rintf(kx * 2.0f);     // jnp.round == round-half-even
    const float muy = rintf(ky * 2.0f);
    const float r3  = 3.0f * kSigma;
    const bool oob = (mux - r3 >= (float)kW) || (muy - r3 >= (float)kW) ||
                     (mux + r3 + 1.0f < 0.0f) || (muy + r3 + 1.0f < 0.0f);
    const bool visible = vis >= 0.5f;
    const float w = (visible && oob) ? 0.0f : vis;

    if (visible && !oob && w != 0.0f) {     // skip -> saves ~20% HBM traffic
      const float mu = half ? muy : mux;

      // ---- async DMA: 2 KB row -> this wave's private LDS buffer ---------
      // 4 x global_load_async_to_lds_b128: 32 lanes x 16 B = 512 B each.
      const float* gbase = pred + (size_t)row * kRowLen;
      const unsigned ldsbase =
          (unsigned)(unsigned long long)(uintptr_t)(&buf[wv][0]);
      #pragma unroll
      for (int q = 0; q < 4; ++q) {
        const unsigned ldst = ldsbase + (unsigned)((q * 128 + lane * 4) * 4);
        const float*   ga   = gbase + q * 128 + lane * 4;
        asm volatile("global_load_async_to_lds_b128 %0, %1, off"
                     :: "v"(ldst), "v"(ga) : "memory");
      }
      asm volatile("s_wait_asynccnt 0" ::: "memory");

      // ---- LDS -> registers: single read pass, row stays resident --------
      float4 rv[4];
      #pragma unroll
      for (int q = 0; q < 4; ++q)
        rv[q] = *reinterpret_cast<const float4*>(&buf[wv][q * 128 + lane * 4]);

      // ---- pass 1: row max of pred (for stable log-softmax) --------------
      float mp = rv[0].x;
      #pragma unroll
      for (int q = 0; q < 4; ++q)
        mp = fmaxf(mp, fmaxf(fmaxf(rv[q].x, rv[q].y), fmaxf(rv[q].z, rv[q].w)));
      #pragma unroll
      for (int m = 16; m >= 1; m >>= 1)
        mp = fmaxf(mp, __shfl_xor(mp, m, 32));
      const float mscale = kBeta * mp;                  // m = max(10*p)

      // ---- pass 2 (registers only): Sp, St, A -----------------------------
      // KL*512 = A/St + m + log(Sp) - log(St), with
      //   Sp = sum exp(10p - m), St = sum exp(10t), A = sum exp(10t)(10t-10p)
      const float inv2s2 = 1.0f / (2.0f * kSigma * kSigma);
      const float jbase  = (float)(lane * 4);
      float sp = 0.0f, st = 0.0f, aa = 0.0f;
      #pragma unroll
      for (int q = 0; q < 4; ++q) {
        const float pe[4] = {rv[q].x, rv[q].y, rv[q].z, rv[q].w};
        #pragma unroll
        for (int e = 0; e < 4; ++e) {
          const float qv = kBeta * pe[e];               // 10*p_j
          const float j  = jbase + (float)(q * 128 + e);
          const float d  = j - mu;
          const float tg = kBeta * __expf(-d * d * inv2s2);  // 10*t_j
          const float et = __expf(tg);                  // exp(10*t_j)
          sp += __expf(qv - mscale);
          st += et;
          aa  = fmaf(et, tg - qv, aa);
        }
      }
      #pragma unroll
      for (int m = 16; m >= 1; m >>= 1) {
        sp += __shfl_xor(sp, m, 32);
        st += __shfl_xor(st, m, 32);
        aa += __shfl_xor(aa, m, 32);
      }
      rowres = w * (aa / st + mscale + __logf(sp) - __logf(st)) *
               (1.0f / (float)kRowLen);
    }
  }

  if (lane == 0) acc[wv] = rowres;
  __syncthreads();
  if (threadIdx.x == 0) {
    float s = 0.0f;
    #pragma unroll
    for (int i = 0; i < kWavesPerBlk; ++i) s += acc[i];
    part[blockIdx.x] = s;
  }
}

// Deterministic final reduction of the 8512 per-block partials.
__global__ __launch_bounds__(256) void rtmcc_reduce_kernel(
    const float* __restrict__ part, int n, float* __restrict__ out)
{
  __shared__ float red[256];
  float s = 0.0f;
  for (int i = threadIdx.x; i < n; i += 256) s += part[i];
  red[threadIdx.x] = s;
  __syncthreads();
  #pragma unroll
  for (int off = 128; off > 0; off >>= 1) {
    if ((int)threadIdx.x < off) red[threadIdx.x] += red[threadIdx.x + off];
    __syncthreads();
  }
  if (threadIdx.x == 0) out[0] = red[0] * (1.0f / (float)kK);
}

extern "C" void kernel_launch(void* const* d_in, const int* in_sizes, int n_in,
                              void* d_out, int out_size, void* d_ws, size_t ws_size,
                              hipStream_t stream) {
  const float* pred = (const float*)d_in[0];   // (256,133,1024) f32
  const float* kp   = (const float*)d_in[1];   // (256,133,3)    f32
  float* out  = (float*)d_out;                 // scalar f32
  float* part = (float*)d_ws;                  // 8512 f32 partials (~34 KB)

  rtmcc_row_kernel<<<kBlocks, 256, 0, stream>>>(pred, kp, part);
  rtmcc_reduce_kernel<<<1, 256, 0, stream>>>(part, kBlocks, out);
}